// Model_74620761801088
// MI455X (gfx1250) — compile-verified
//
#include <hip/hip_runtime.h>
#include <hip/hip_bf16.h>

typedef __attribute__((ext_vector_type(2))) float v2f;
typedef __attribute__((ext_vector_type(8))) float v8f;

#define N_NODES 50000
#define N_EDGES 800000
#define IN_DIM  128
#define HIDDEN  32
#define HEADS   8
#define OUT_DIM 64
#define NEG_SLOPE 0.2f

// ---------------------------------------------------------------------------
// Generic float fill (init accumulators / -inf max buffers each call)
// ---------------------------------------------------------------------------
__global__ void fill_f32(float* __restrict__ p, float v, int n) {
    int i = blockIdx.x * blockDim.x + threadIdx.x;
    if (i < n) p[i] = v;
}

// ---------------------------------------------------------------------------
// C[M,N] = A[M,K] * B[K,N] in full fp32 using V_WMMA_F32_16X16X4_F32.
// One wave32 computes one 16x16 tile of C; K-loop in steps of 4.
// Requires M%16==0, N%16==0, K%4==0, blockDim.x%32==0.
// A-operand (16x4, 2 VGPRs): lane l holds row (l&15), K = 2*(l>>4)+{0,1}
// B-operand (4x16, 2 VGPRs): lane l holds col (l&15), K = 2*(l>>4)+{0,1}
// C/D layout (8 VGPRs): acc[r] -> row 8*(l>>4)+r, col (l&15)
// ---------------------------------------------------------------------------
__global__ void gemm_f32_wmma(const float* __restrict__ A,
                              const float* __restrict__ B,
                              float* __restrict__ C,
                              int M, int N, int K) {
    const int lane = threadIdx.x & 31;
    const int wave = threadIdx.x >> 5;
    const int tiles_n = N >> 4;
    const int tile = blockIdx.x * (blockDim.x >> 5) + wave;
    const int total = (M >> 4) * tiles_n;
    if (tile >= total) return;            // wave-uniform: EXEC stays all-1s

    const int tm  = tile / tiles_n;
    const int tn  = tile % tiles_n;
    const int r15 = lane & 15;
    const int hi  = lane >> 4;            // 0 or 1
    const long arow = (long)(tm * 16 + r15) * K;
    const int  ncol = tn * 16 + r15;
    const int  koff = hi * 2;

    v8f acc = {};
    for (int k = 0; k < K; k += 4) {
        v2f a = *(const v2f*)(A + arow + k + koff);   // contiguous 8B load
        v2f b;
        b.x = B[(long)(k + koff    ) * N + ncol];
        b.y = B[(long)(k + koff + 1) * N + ncol];
        acc = __builtin_amdgcn_wmma_f32_16x16x4_f32(
            /*neg_a=*/false, a, /*neg_b=*/false, b,
            /*c_mod=*/(short)0, acc, /*reuse_a=*/false, /*reuse_b=*/false);
    }

    float* cp = C + (long)(tm * 16 + hi * 8) * N + ncol;
#pragma unroll
    for (int r = 0; r < 8; ++r) cp[(long)r * N] = acc[r];
}

// ---------------------------------------------------------------------------
// el[n,h] = dot(f[n,h,:], al[h,:]) ; er likewise. One thread per (node,head).
// H, D compile-time -> % and * become masks/shifts.
// ---------------------------------------------------------------------------
template <int H, int D>
__global__ void attn_halves(const float* __restrict__ f,
                            const float* __restrict__ al,
                            const float* __restrict__ ar,
                            float* __restrict__ el,
                            float* __restrict__ er,
                            int nh) {
    int idx = blockIdx.x * blockDim.x + threadIdx.x;
    if (idx >= nh) return;
    int h = idx % H;                 // H power of two -> mask
    const float* fp  = f  + (long)idx * D;
    const float* alp = al + h * D;
    const float* arp = ar + h * D;
    float sl = 0.f, sr = 0.f;
#pragma unroll 4
    for (int i = 0; i < D; ++i) {
        float v = fp[i];
        sl = fmaf(v, alp[i], sl);
        sr = fmaf(v, arp[i], sr);
    }
    el[idx] = sl;
    er[idx] = sr;
}

__device__ __forceinline__ float leaky(float x) {
    return x > 0.f ? x : NEG_SLOPE * x;
}

// Segment max over incoming edges (GLOBAL_ATOMIC_MAX_NUM_F32 path)
template <int H>
__global__ void edge_max(const int* __restrict__ src, const int* __restrict__ dst,
                         const float* __restrict__ el, const float* __restrict__ er,
                         float* __restrict__ m, int EH) {
    int idx = blockIdx.x * blockDim.x + threadIdx.x;
    if (idx >= EH) return;
    int e = idx / H;                 // shift
    int h = idx % H;                 // mask
    int s = src[e], d = dst[e];
    float x = leaky(el[s * H + h] + er[d * H + h]);
    atomicMax(&m[d * H + h], x);
}

// Segment sum of exp(e - m[dst])
template <int H>
__global__ void edge_expsum(const int* __restrict__ src, const int* __restrict__ dst,
                            const float* __restrict__ el, const float* __restrict__ er,
                            const float* __restrict__ m, float* __restrict__ sb,
                            int EH) {
    int idx = blockIdx.x * blockDim.x + threadIdx.x;
    if (idx >= EH) return;
    int e = idx / H;
    int h = idx % H;
    int s = src[e], d = dst[e];
    int dh = d * H + h;
    float x = leaky(el[s * H + h] + er[dh]);
    atomicAdd(&sb[dh], __expf(x - m[dh]));
}

// alpha[e,h] = exp(e - m) / (s>0 ? s : 1)
template <int H>
__global__ void edge_alpha(const int* __restrict__ src, const int* __restrict__ dst,
                           const float* __restrict__ el, const float* __restrict__ er,
                           const float* __restrict__ m, const float* __restrict__ sb,
                           float* __restrict__ alpha, int EH) {
    int idx = blockIdx.x * blockDim.x + threadIdx.x;
    if (idx >= EH) return;
    int e = idx / H;
    int h = idx % H;
    int s = src[e], d = dst[e];
    int dh = d * H + h;
    float x  = leaky(el[s * H + h] + er[dh]);
    float sv = sb[dh];
    alpha[idx] = __expf(x - m[dh]) / (sv > 0.f ? sv : 1.f);
}

// out[dst, h, dch] += f[src, h, dch] * alpha[e, h]
// Consecutive threads cover consecutive channels of one edge -> coalesced
// gather of f[src] and coalesced float atomic-adds into out[dst].
// H*D is a power of two -> pure shift/mask index decomposition.
template <int H, int D>
__global__ void edge_message(const int* __restrict__ src, const int* __restrict__ dst,
                             const float* __restrict__ f, const float* __restrict__ alpha,
                             float* __restrict__ out, int total) {
    int gid = blockIdx.x * blockDim.x + threadIdx.x;
    if (gid >= total) return;
    constexpr int HD = H * D;
    int e = gid / HD;                // shift
    int c = gid % HD;                // mask
    float a = alpha[e * H + c / D];  // shift
    int s = src[e], d = dst[e];
    atomicAdd(&out[d * HD + c], f[s * HD + c] * a);
}

// x = in + bias ; out = elu(x)
template <int C>
__global__ void bias_elu(const float* __restrict__ in, const float* __restrict__ b,
                         float* __restrict__ out, int total) {
    int i = blockIdx.x * blockDim.x + threadIdx.x;
    if (i >= total) return;
    float x = in[i] + b[i % C];      // mask
    out[i] = x > 0.f ? x : expm1f(x);
}

// out = in + bias (final output; heads==1 so mean over axis 1 is identity)
template <int C>
__global__ void bias_add(const float* __restrict__ in, const float* __restrict__ b,
                         float* __restrict__ out, int total) {
    int i = blockIdx.x * blockDim.x + threadIdx.x;
    if (i >= total) return;
    out[i] = in[i] + b[i % C];
}

// ---------------------------------------------------------------------------
static inline int nblk(long total, int bs) { return (int)((total + bs - 1) / bs); }

extern "C" void kernel_launch(void* const* d_in, const int* in_sizes, int n_in,
                              void* d_out, int out_size, void* d_ws, size_t ws_size,
                              hipStream_t stream) {
    const float* feat = (const float*)d_in[0];
    const int*   src  = (const int*)  d_in[1];
    const int*   dst  = (const int*)  d_in[2];
    const float* W1   = (const float*)d_in[3];
    const float* al1  = (const float*)d_in[4];
    const float* ar1  = (const float*)d_in[5];
    const float* b1   = (const float*)d_in[6];
    const float* W2   = (const float*)d_in[7];
    const float* al2  = (const float*)d_in[8];
    const float* ar2  = (const float*)d_in[9];
    const float* b2   = (const float*)d_in[10];
    float* out = (float*)d_out;
    (void)in_sizes; (void)n_in; (void)out_size; (void)ws_size;

    const int FD1 = HEADS * HIDDEN;   // 256
    float* w = (float*)d_ws;
    float* f1     = w; w += (long)N_NODES * FD1;      // projected feats L1
    float* el1    = w; w += (long)N_NODES * HEADS;
    float* er1    = w; w += (long)N_NODES * HEADS;
    float* m1     = w; w += (long)N_NODES * HEADS;
    float* s1     = w; w += (long)N_NODES * HEADS;
    float* alpha1 = w; w += (long)N_EDGES * HEADS;
    float* h1     = w; w += (long)N_NODES * FD1;      // aggregated -> elu'd
    float* f2     = w; w += (long)N_NODES * OUT_DIM;  // projected feats L2
    float* el2    = w; w += (long)N_NODES;
    float* er2    = w; w += (long)N_NODES;
    float* m2     = w; w += (long)N_NODES;
    float* s2     = w; w += (long)N_NODES;
    float* alpha2 = w; w += (long)N_EDGES;
    float* o2     = w; w += (long)N_NODES * OUT_DIM;  // aggregated L2

    const int BS = 256;
    const float NEG_INF = -__builtin_huge_valf();

    // ---- init accumulators (deterministic every call) ----
    int nNH = N_NODES * HEADS;
    fill_f32<<<nblk(nNH, BS), BS, 0, stream>>>(m1, NEG_INF, nNH);
    fill_f32<<<nblk(nNH, BS), BS, 0, stream>>>(s1, 0.f, nNH);
    fill_f32<<<nblk(N_NODES * FD1, BS), BS, 0, stream>>>(h1, 0.f, N_NODES * FD1);
    fill_f32<<<nblk(N_NODES, BS), BS, 0, stream>>>(m2, NEG_INF, N_NODES);
    fill_f32<<<nblk(N_NODES, BS), BS, 0, stream>>>(s2, 0.f, N_NODES);
    fill_f32<<<nblk(N_NODES * OUT_DIM, BS), BS, 0, stream>>>(o2, 0.f, N_NODES * OUT_DIM);

    // ---- Layer 1 ----
    // f1 = feat @ W1   (M=50000, N=256, K=128), 8 WMMA waves per block
    {
        long tiles = (long)(N_NODES / 16) * (FD1 / 16);
        gemm_f32_wmma<<<nblk(tiles, 8), 256, 0, stream>>>(feat, W1, f1, N_NODES, FD1, IN_DIM);
    }
    attn_halves<HEADS, HIDDEN><<<nblk(nNH, BS), BS, 0, stream>>>(f1, al1, ar1, el1, er1, nNH);

    int EH1 = N_EDGES * HEADS;
    edge_max<HEADS>   <<<nblk(EH1, BS), BS, 0, stream>>>(src, dst, el1, er1, m1, EH1);
    edge_expsum<HEADS><<<nblk(EH1, BS), BS, 0, stream>>>(src, dst, el1, er1, m1, s1, EH1);
    edge_alpha<HEADS> <<<nblk(EH1, BS), BS, 0, stream>>>(src, dst, el1, er1, m1, s1, alpha1, EH1);

    int msg1 = N_EDGES * FD1;   // 204.8M lanes, coalesced gather+scatter
    edge_message<HEADS, HIDDEN><<<nblk(msg1, BS), BS, 0, stream>>>(src, dst, f1, alpha1, h1, msg1);

    bias_elu<FD1><<<nblk(N_NODES * FD1, BS), BS, 0, stream>>>(h1, b1, h1, N_NODES * FD1);

    // ---- Layer 2 ----
    // f2 = h1 @ W2   (M=50000, N=64, K=256)
    {
        long tiles = (long)(N_NODES / 16) * (OUT_DIM / 16);
        gemm_f32_wmma<<<nblk(tiles, 8), 256, 0, stream>>>(h1, W2, f2, N_NODES, OUT_DIM, FD1);
    }
    attn_halves<1, OUT_DIM><<<nblk(N_NODES, BS), BS, 0, stream>>>(f2, al2, ar2, el2, er2, N_NODES);

    int EH2 = N_EDGES;
    edge_max<1>   <<<nblk(EH2, BS), BS, 0, stream>>>(src, dst, el2, er2, m2, EH2);
    edge_expsum<1><<<nblk(EH2, BS), BS, 0, stream>>>(src, dst, el2, er2, m2, s2, EH2);
    edge_alpha<1> <<<nblk(EH2, BS), BS, 0, stream>>>(src, dst, el2, er2, m2, s2, alpha2, EH2);

    int msg2 = N_EDGES * OUT_DIM;
    edge_message<1, OUT_DIM><<<nblk(msg2, BS), BS, 0, stream>>>(src, dst, f2, alpha2, o2, msg2);

    // final: out = o2 + b2 (mean over the single head is a no-op)
    bias_add<OUT_DIM><<<nblk(N_NODES * OUT_DIM, BS), BS, 0, stream>>>(o2, b2, out, N_NODES * OUT_DIM);
}